// OutputBlock_18562848654098
// MI455X (gfx1250) — compile-verified
//
#include <hip/hip_runtime.h>
#include <hip/hip_bf16.h>
#include <math.h>

// CDNA5 / gfx1250, wave32. fp32 end-to-end; matrix ops on V_WMMA_F32_16X16X4_F32,
// data staging via GLOBAL_LOAD_ASYNC_TO_LDS_B128 (ASYNCcnt), NT hints on the
// single-use edge streams.

typedef __attribute__((ext_vector_type(2))) float v2f;
typedef __attribute__((ext_vector_type(4))) float v4f;
typedef __attribute__((ext_vector_type(8))) float v8f;

#define WMMA_F32(a, b, c) \
    __builtin_amdgcn_wmma_f32_16x16x4_f32(false, (a), false, (b), (short)0, (c), false, false)

__device__ __forceinline__ float silu(float z) {
    return z / (1.0f + __expf(-z));
}

// LDS byte address of a generic pointer into shared memory (flat LDS aperture
// stores the LDS byte offset in addr[31:0]).
__device__ __forceinline__ unsigned lds_addr(const void* p) {
    return (unsigned)(unsigned long long)p;
}

// Async copy 16B global -> LDS (tracked by ASYNCcnt, not LOADcnt).
__device__ __forceinline__ void async_b128(unsigned lds_off, const float* g) {
    asm volatile("global_load_async_to_lds_b128 %0, %1, off"
                 :: "v"(lds_off), "v"(g) : "memory");
}
__device__ __forceinline__ void wait_async0() {
    asm volatile("s_wait_asynccnt 0" ::: "memory");
}
__device__ __forceinline__ void fence_ds() {
    asm volatile("s_wait_dscnt 0" ::: "memory");
}

// ---------------------------------------------------------------------------
// Kernel 0: zero the node accumulator (b128 stores).
// ---------------------------------------------------------------------------
__global__ void zero_kernel(v4f* __restrict__ p, unsigned long long n4) {
    unsigned long long i = (unsigned long long)blockIdx.x * blockDim.x + threadIdx.x;
    if (i < n4) p[i] = (v4f){0.f, 0.f, 0.f, 0.f};
}

// ---------------------------------------------------------------------------
// Kernel 1: edge = (pair_basis @ W_pair) * x ; atomic scatter-add into node[].
// One wave: 16 edges x 128 hidden, 32 f32 WMMAs, fully unrolled so all x-loads
// and atomics use one base pointer + immediate offsets. x / pair_basis are
// single-use streams (~370MB => ~16us floor at 23.3 TB/s) -> NT loads.
// Atomics: lanes 0-15 / 16-31 of each instruction hit one 64B line each of a
// node row, so the scatter is line-coalesced into L2 (node array 25.6MB).
// ---------------------------------------------------------------------------
template <bool FULL>
__device__ __forceinline__ void edge_tile_body(
    const float* __restrict__ x, const float* __restrict__ Wp,
    float* __restrict__ node, const v2f (&a)[4], const int (&ni)[8],
    int e0, int nEdges, int hi, int l15)
{
    // Base pointers: all subsequent addressing is immediate offsets.
    const float* xb  = x  + (size_t)(e0 + 8 * hi) * 128 + l15;  // + v*128 + n*16
    const float* wpb = Wp + 2 * hi * 128 + l15;                 // + (4kk)*128 + n*16
    float* nb[8];
    #pragma unroll
    for (int v = 0; v < 8; ++v) nb[v] = node + (size_t)ni[v] * 128 + l15;

    #pragma unroll
    for (int n = 0; n < 8; ++n) {
        v8f c = {};
        #pragma unroll
        for (int kk = 0; kk < 4; ++kk) {
            v2f b;
            b.x = wpb[(4 * kk + 0) * 128 + n * 16];
            b.y = wpb[(4 * kk + 1) * 128 + n * 16];
            c = WMMA_F32(a[kk], b, c);
        }
        // Phase 1: clause of 8 NT x-loads (immediate offsets).
        float xv[8];
        #pragma unroll
        for (int v = 0; v < 8; ++v)
            if (FULL || (e0 + v + 8 * hi < nEdges))
                xv[v] = __builtin_nontemporal_load(xb + v * 128 + n * 16);
        // Phase 2: clause of 8 gate+atomic ops.
        #pragma unroll
        for (int v = 0; v < 8; ++v)
            if (FULL || (e0 + v + 8 * hi < nEdges))
                atomicAdd(nb[v] + n * 16, c[v] * xv[v]);
    }
}

__global__ __launch_bounds__(256) void edge_scatter_kernel(
    const float* __restrict__ x,      // [E,128]
    const float* __restrict__ pb,     // [E,16]
    const int*   __restrict__ idx,    // [E]
    const float* __restrict__ Wp,     // [16,128]
    float* __restrict__ node,         // [N,128] accumulator
    int nEdges)
{
    const int lane = threadIdx.x & 31;
    const int wid  = threadIdx.x >> 5;
    const int tile = blockIdx.x * 8 + wid;           // 8 waves / block
    const int e0   = tile * 16;
    if (e0 >= nEdges) return;                        // uniform per wave

    const int hi  = lane >> 4;                       // half-wave: 0 or 1
    const int l15 = lane & 15;

    // A fragments: 16x16 pair_basis tile as four 16x4 slabs
    // (lane m holds row m; halves hold K = 2*hi + {0,1} per slab). NT stream.
    v2f a[4];
    {
        int erow = e0 + l15; if (erow >= nEdges) erow = nEdges - 1;
        const float* prow = pb + (size_t)erow * 16 + 2 * hi;
        #pragma unroll
        for (int kk = 0; kk < 4; ++kk) {
            a[kk].x = __builtin_nontemporal_load(prow + 4 * kk + 0);
            a[kk].y = __builtin_nontemporal_load(prow + 4 * kk + 1);
        }
    }

    // Destination node indices for the 8 C-rows this lane covers.
    int ni[8];
    #pragma unroll
    for (int v = 0; v < 8; ++v) {
        int erow = e0 + v + 8 * hi; if (erow >= nEdges) erow = nEdges - 1;
        ni[v] = idx[erow];
    }

    if (e0 + 16 <= nEdges)   // wave-uniform: no per-element exec masking
        edge_tile_body<true >(x, Wp, node, a, ni, e0, nEdges, hi, l15);
    else
        edge_tile_body<false>(x, Wp, node, a, ni, e0, nEdges, hi, l15);
}

// ---------------------------------------------------------------------------
// Kernel 2: fused MLP  out = (silu(silu(node@W1)@W2 + b2)) @ W3
// Block = 4 waves x 16 nodes. Weight column-slabs (Kx16 f32) staged
// block-cooperatively into double-buffered LDS via global_load_async_to_lds,
// overlapping the async DMA of slab n+1 with the 32/64 chained WMMAs of slab
// n, and sharing each slab across 4 waves (4x less L2 weight traffic). Node
// tiles async-staged as coalesced 512B rows. LDS strides padded (132 / 260
// floats) so A/B fragment ds_loads are bank-conflict-free; A fragments load
// as single ds_load_b64.
// ---------------------------------------------------------------------------
#define HSTR 260                 // padded row stride of per-wave h buffer
#define NSTR 132                 // padded row stride of staged node tile
#define MLP_LDS_BYTES (4 * 16 * HSTR * 4 + 2 * 256 * 16 * 4)   // 99328 B

__device__ __forceinline__ void stage_slab_async(const float* __restrict__ W,
                                                 int K, int n, float* dst, int tid) {
    const int r = tid >> 2;            // 0..31 : row within pass
    const int q = (tid & 3) * 4;       // column quad (4 floats = 16B)
    #pragma unroll
    for (int p = 0; p < 256; p += 32) {
        if (p < K)
            async_b128(lds_addr(dst + (p + r) * 16 + q),
                       W + (size_t)(p + r) * 256 + n * 16 + q);
    }
}

__global__ __launch_bounds__(128) void node_mlp_kernel(
    const float* __restrict__ node,   // [N,128]
    const float* __restrict__ W1,     // [128,256]
    const float* __restrict__ W2,     // [256,256]
    const float* __restrict__ b2,     // [256]
    const float* __restrict__ W3,     // [256]
    float* __restrict__ out,          // [N]
    int nNodes)
{
    extern __shared__ float smem[];
    const int tid  = threadIdx.x;
    const int lane = tid & 31;
    const int wid  = tid >> 5;
    const int hi   = lane >> 4;
    const int l15  = lane & 15;

    float* hbuf   = smem + wid * (16 * HSTR);   // per-wave 16x256 (padded)
    float* wslab0 = smem + 4 * (16 * HSTR);     // shared 256x16 slab, buf 0
    float* wslab1 = wslab0 + 256 * 16;          // buf 1

    const int numTiles = (nNodes + 15) / 16;
    int tile = blockIdx.x * 4 + wid;
    const bool active = (tile < numTiles);
    if (!active) tile = numTiles - 1;           // clamp: EXEC stays all-1s
    const int r0 = tile * 16;

    // ---- Stage this wave's 16x128 node tile into LDS (async, coalesced:
    //      each row is one 512B burst = 32 lanes x b128).
    #pragma unroll
    for (int r = 0; r < 16; ++r) {
        int rr = r0 + r; if (rr >= nNodes) rr = nNodes - 1;
        async_b128(lds_addr(hbuf + r * NSTR + lane * 4),
                   node + (size_t)rr * 128 + lane * 4);
    }
    wait_async0();                              // wave-local data, no barrier

    // ---- A1 fragments from LDS (b64 loads, conflict-free: stride 132).
    v2f a1[32];
    #pragma unroll
    for (int kk = 0; kk < 32; ++kk)
        a1[kk] = *(const v2f*)(hbuf + l15 * NSTR + 4 * kk + 2 * hi);
    fence_ds();                                 // a1 reads before hbuf reuse

    // ---- Layer 1: h1 = silu(A @ W1), 16 col tiles x 32 WMMAs, slab double
    //      buffered: async-stage tile n+1 while computing tile n.
    stage_slab_async(W1, 128, 0, wslab0, tid);
    #pragma unroll 1
    for (int n = 0; n < 16; ++n) {
        float* cur = (n & 1) ? wslab1 : wslab0;
        float* nxt = (n & 1) ? wslab0 : wslab1;
        wait_async0();
        __syncthreads();                        // slab n visible to all waves
        if (n + 1 < 16) stage_slab_async(W1, 128, n + 1, nxt, tid);
        v8f c = {};
        #pragma unroll
        for (int kk = 0; kk < 32; ++kk) {
            v2f b;
            const int k = 4 * kk + 2 * hi;
            b.x = cur[(k + 0) * 16 + l15];
            b.y = cur[(k + 1) * 16 + l15];
            c = WMMA_F32(a1[kk], b, c);
        }
        #pragma unroll
        for (int v = 0; v < 8; ++v)
            hbuf[(v + 8 * hi) * HSTR + n * 16 + l15] = silu(c[v]);
    }
    fence_ds();

    // ---- Relayout h1 into A2 fragments (b64 loads; DS in-order per wave).
    v2f a2[64];
    #pragma unroll
    for (int kk = 0; kk < 64; ++kk)
        a2[kk] = *(const v2f*)(hbuf + l15 * HSTR + 4 * kk + 2 * hi);
    fence_ds();

    // ---- Layer 2: h2 = silu(h1 @ W2 + b2), 16 col tiles x 64 WMMAs.
    stage_slab_async(W2, 256, 0, wslab0, tid);
    #pragma unroll 1
    for (int n = 0; n < 16; ++n) {
        float* cur = (n & 1) ? wslab1 : wslab0;
        float* nxt = (n & 1) ? wslab0 : wslab1;
        wait_async0();
        __syncthreads();
        if (n + 1 < 16) stage_slab_async(W2, 256, n + 1, nxt, tid);
        v8f c = {};
        #pragma unroll
        for (int kk = 0; kk < 64; ++kk) {
            v2f b;
            const int k = 4 * kk + 2 * hi;
            b.x = cur[(k + 0) * 16 + l15];
            b.y = cur[(k + 1) * 16 + l15];
            c = WMMA_F32(a2[kk], b, c);
        }
        const float bias = b2[n * 16 + l15];
        #pragma unroll
        for (int v = 0; v < 8; ++v)
            hbuf[(v + 8 * hi) * HSTR + n * 16 + l15] = silu(c[v] + bias);
    }
    fence_ds();

    // ---- Layer 3: out[r] = h2[r,:] . W3. Lane l15 owns row l15; halves
    //      split the 256 cols, then pair-reduce across the half-waves.
    float acc = 0.0f;
    {
        const float* hrow = hbuf + l15 * HSTR + hi * 128;
        const float* w3   = W3 + hi * 128;
        #pragma unroll 8
        for (int k = 0; k < 128; ++k)
            acc = fmaf(hrow[k], w3[k], acc);
    }
    acc += __shfl_xor(acc, 16, 32);
    if (active && lane < 16 && (r0 + l15) < nNodes)
        __builtin_nontemporal_store(acc, &out[r0 + l15]);
}

// ---------------------------------------------------------------------------
// Host-side launcher. Inputs (setup_inputs order):
//   0: x[E,128] f32   1: pair_basis[E,16] f32   2: i[E] int
//   3: num_nodes (scalar)   4: W_pair[16,128]   5: W1[128,256]
//   6: W2[256,256]   7: b2[256]   8: W3[256,1]
// d_out: [N] f32.  d_ws: node accumulator [N,128] f32 (25.6 MB).
// ---------------------------------------------------------------------------
extern "C" void kernel_launch(void* const* d_in, const int* in_sizes, int n_in,
                              void* d_out, int out_size, void* d_ws, size_t ws_size,
                              hipStream_t stream) {
    const float* x   = (const float*)d_in[0];
    const float* pb  = (const float*)d_in[1];
    const int*   idx = (const int*)d_in[2];
    const float* Wp  = (const float*)d_in[4];
    const float* W1  = (const float*)d_in[5];
    const float* W2  = (const float*)d_in[6];
    const float* b2  = (const float*)d_in[7];
    const float* W3  = (const float*)d_in[8];
    float* out = (float*)d_out;

    const int E = in_sizes[0] / 128;       // 640000
    const int N = out_size;                // 50000 (OUT_CH == 1)
    float* node = (float*)d_ws;
    const unsigned long long node4 = (unsigned long long)N * 32ull;  // float4 count

    // 0) zero the accumulator (b128 stores)
    zero_kernel<<<(unsigned)((node4 + 255) / 256), 256, 0, stream>>>((v4f*)node, node4);

    // 1) edge transform + scatter (16 edges/wave, 8 waves/block)
    const int eTiles = (E + 15) / 16;
    edge_scatter_kernel<<<(eTiles + 7) / 8, 256, 0, stream>>>(x, pb, idx, Wp, node, E);

    // 2) fused node MLP (16 nodes/wave, 4 waves/block, 99KB dynamic LDS)
    const int nTiles = (N + 15) / 16;
    node_mlp_kernel<<<(nTiles + 3) / 4, 128, MLP_LDS_BYTES, stream>>>(
        node, W1, W2, b2, W3, out, N);
}